// GNN_1_EFGS_75986561401174
// MI455X (gfx1250) — compile-verified
//
#include <hip/hip_runtime.h>

// ---------------- problem constants (match reference setup_inputs) ----------
static constexpr int   NN   = 100000;   // nodes
static constexpr int   EE   = 1600000;  // edges
static constexpr int   N3   = 20000;    // clusters
static constexpr int   AA   = 100000;   // assignment entries
static constexpr int   E3   = 100000;   // cluster edges
static constexpr int   GG   = 2000;     // graphs
static constexpr int   FIN  = 128;
static constexpr int   EMB  = 256;
static constexpr int   VV   = 208;
static constexpr int   LL   = 5;
static constexpr float INV_BN = 0.9999950000374996f; // 1/sqrt(1+1e-5)

typedef float v2f __attribute__((ext_vector_type(2)));
typedef float v8f __attribute__((ext_vector_type(8)));

// ---------------- WMMA f32 GEMM: C[M,NC] = A[M,K] @ W[K,NC] (+bias)(+Cadd)(relu)
// One wave owns a full 16xNC strip: NC/16 independent accumulators, so the
// A fragment is loaded ONCE per K-step (read-once A traffic) and the WMMA
// issues form NC/16 independent dependency chains (hides XDL latency).
// Requires: M%16==0 handled by wave-uniform early-out, K%4==0, NC in {64,128,256}.
template <int NC>
__global__ void __launch_bounds__(128)
gemm_wmma_f32_nc(const float* __restrict__ A, const float* __restrict__ W,
                 const float* __restrict__ bias, const float* __restrict__ Cadd,
                 float* __restrict__ C, int M, int K, int relu)
{
    constexpr int NT = NC / 16;
    const int lane = threadIdx.x & 31;
    const int wave = threadIdx.x >> 5;
    const int m0 = (blockIdx.x * 4 + wave) * 16;
    if (m0 >= M) return;                         // wave-uniform

    const int l  = lane & 15;
    const int kh = (lane >> 4) << 1;             // 0 for lanes 0-15, 2 for 16-31

    v8f acc[NT];
    #pragma unroll
    for (int t = 0; t < NT; ++t) acc[t] = (v8f){0.f,0.f,0.f,0.f,0.f,0.f,0.f,0.f};

    const float* arow = A + (size_t)(m0 + l) * K + kh;   // K=kh, kh+1
    const float* brow = W + (size_t)kh * NC + l;         // rows kh, kh+1, col base l

    for (int k = 0; k < K; k += 4) {
        const float2 av = *(const float2*)arow;          // 8B-aligned A frag
        v2f a; a[0] = av.x; a[1] = av.y;
        #pragma unroll
        for (int t = 0; t < NT; ++t) {
            v2f b;
            b[0] = brow[t * 16];
            b[1] = brow[t * 16 + NC];
            acc[t] = __builtin_amdgcn_wmma_f32_16x16x4_f32(
                         /*neg_a=*/false, a, /*neg_b=*/false, b,
                         /*c_mod=*/(short)0, acc[t],
                         /*reuse_a=*/false, /*reuse_b=*/false);
        }
        arow += 4;
        brow += (size_t)4 * NC;
    }

    const int rbase = m0 + ((lane >> 4) << 3);           // +8 for upper half-wave
    #pragma unroll
    for (int t = 0; t < NT; ++t) {
        const int col = t * 16 + l;
        const float bv = bias ? bias[col] : 0.f;
        #pragma unroll
        for (int r = 0; r < 8; ++r) {
            const int row = rbase + r;
            float v = acc[t][r] + bv;
            if (Cadd) v += Cadd[(size_t)row * NC + col];
            if (relu) v = fmaxf(v, 0.f);
            C[(size_t)row * NC + col] = v;
        }
    }
}

// ---------------- elementwise / scatter helpers ----------------------------
__global__ void fill_f32(float* __restrict__ p, float v, long long n)
{
    long long i = (long long)blockIdx.x * blockDim.x + threadIdx.x;
    if (i < n) p[i] = v;
}

__global__ void deg_accum(const int* __restrict__ dst, float* __restrict__ deg, int E)
{
    int e = blockIdx.x * blockDim.x + threadIdx.x;
    if (e < E) atomicAdd(&deg[dst[e]], 1.0f);
}

__global__ void rsqrt_inplace(float* __restrict__ d, int n)
{
    int i = blockIdx.x * blockDim.x + threadIdx.x;
    if (i < n) d[i] = rsqrtf(fmaxf(d[i], 1.0f));
}

__global__ void edge_norm_k(const int* __restrict__ src, const int* __restrict__ dst,
                            const float* __restrict__ dinv, float* __restrict__ nrm, int E)
{
    int e = blockIdx.x * blockDim.x + threadIdx.x;
    if (e < E) nrm[e] = dinv[src[e]] * dinv[dst[e]];
}

// agg[dst[e], :] += x[src[e], :] * (nrm ? nrm[e] : 1)   (float4 chunks)
__global__ void scatter_edges(const float* __restrict__ x, const int* __restrict__ src,
                              const int* __restrict__ dst, const float* __restrict__ nrm,
                              float* __restrict__ agg, long long total, int Fq, int F)
{
    long long idx = (long long)blockIdx.x * blockDim.x + threadIdx.x;
    if (idx >= total) return;
    const int e = (int)(idx / Fq);
    const int f = (int)(idx % Fq) * 4;
    const int s = src[e], d = dst[e];
    const float w = nrm ? nrm[e] : 1.0f;
    const float4 v = *(const float4*)(x + (size_t)s * F + f);
    float* o = agg + (size_t)d * F + f;
    atomicAdd(o + 0, v.x * w);
    atomicAdd(o + 1, v.y * w);
    atomicAdd(o + 2, v.z * w);
    atomicAdd(o + 3, v.w * w);
}

// sums[seg[i], :] += x[i, :];  cnt[seg[i]] += 1
__global__ void seg_pool(const float* __restrict__ x, const int* __restrict__ seg,
                         float* __restrict__ sums, float* __restrict__ cnt,
                         long long total, int Fq, int F)
{
    long long idx = (long long)blockIdx.x * blockDim.x + threadIdx.x;
    if (idx >= total) return;
    const int i = (int)(idx / Fq);
    const int f = (int)(idx % Fq) * 4;
    const int g = seg[i];
    const float4 v = *(const float4*)(x + (size_t)i * F + f);
    float* o = sums + (size_t)g * F + f;
    atomicAdd(o + 0, v.x);
    atomicAdd(o + 1, v.y);
    atomicAdd(o + 2, v.z);
    atomicAdd(o + 3, v.w);
    if (f == 0) atomicAdd(&cnt[g], 1.0f);
}

// sums[col[i], :] += x[row[i], :];  cnt[col[i]] += 1
__global__ void seg_pool_gather(const float* __restrict__ x, const int* __restrict__ row,
                                const int* __restrict__ col, float* __restrict__ sums,
                                float* __restrict__ cnt, long long total, int Fq, int F)
{
    long long idx = (long long)blockIdx.x * blockDim.x + threadIdx.x;
    if (idx >= total) return;
    const int i = (int)(idx / Fq);
    const int f = (int)(idx % Fq) * 4;
    const int g = col[i];
    const float4 v = *(const float4*)(x + (size_t)row[i] * F + f);
    float* o = sums + (size_t)g * F + f;
    atomicAdd(o + 0, v.x);
    atomicAdd(o + 1, v.y);
    atomicAdd(o + 2, v.z);
    atomicAdd(o + 3, v.w);
    if (f == 0) atomicAdd(&cnt[g], 1.0f);
}

// dst[i, col0+f] = sums[i, f] / max(cnt[i], 1)
__global__ void mean_to(float* __restrict__ dst, int ldd, int col0,
                        const float* __restrict__ sums, const float* __restrict__ cnt,
                        long long total, int F)
{
    long long idx = (long long)blockIdx.x * blockDim.x + threadIdx.x;
    if (idx >= total) return;
    const int i = (int)(idx / F);
    const int f = (int)(idx % F);
    dst[(size_t)i * ldd + col0 + f] = sums[idx] / fmaxf(cnt[i], 1.0f);
}

// xc = [xp/cnt3 , iso]   (N3 x (EMB+V))
__global__ void build_xc(const float* __restrict__ xp, const float* __restrict__ cnt3,
                         const float* __restrict__ iso, float* __restrict__ xc,
                         long long total)
{
    long long idx = (long long)blockIdx.x * blockDim.x + threadIdx.x;
    if (idx >= total) return;
    const int Fc = EMB + VV;
    const int i = (int)(idx / Fc);
    const int j = (int)(idx % Fc);
    float v;
    if (j < EMB) v = xp[(size_t)i * EMB + j] / fmaxf(cnt3[i], 1.0f);
    else         v = iso[(size_t)i * VV + (j - EMB)];
    xc[idx] = v;
}

// h = BN( agg + hw*dinv^2 + bg + h ), optional relu
__global__ void layer_update(float* __restrict__ h, const float* __restrict__ hw,
                             const float* __restrict__ agg, const float* __restrict__ dinv,
                             const float* __restrict__ bg, const float* __restrict__ gamma,
                             const float* __restrict__ beta, long long total, int F, int relu)
{
    long long idx = (long long)blockIdx.x * blockDim.x + threadIdx.x;
    if (idx >= total) return;
    const int i = (int)(idx / F);
    const int f = (int)(idx % F);
    const float di = dinv[i];
    float s = agg[idx] + hw[idx] * di * di + bg[f] + h[idx];
    s = s * INV_BN * gamma[f] + beta[f];
    if (relu) s = fmaxf(s, 0.f);
    h[idx] = s;
}

// out[g] = bl + dot(m2[g,:], Wl[:,0])
__global__ void final_linear(const float* __restrict__ m2, const float* __restrict__ Wl,
                             const float* __restrict__ bl, float* __restrict__ out,
                             int G, int K)
{
    int g = blockIdx.x * blockDim.x + threadIdx.x;
    if (g >= G) return;
    float acc = bl[0];
    for (int k = 0; k < K; ++k) acc += m2[(size_t)g * K + k] * Wl[k];
    out[g] = acc;
}

// ---------------------------------------------------------------------------
extern "C" void kernel_launch(void* const* d_in, const int* in_sizes, int n_in,
                              void* d_out, int out_size, void* d_ws, size_t ws_size,
                              hipStream_t stream)
{
    const float* x      = (const float*)d_in[0];
    const int*   ei     = (const int*)d_in[1];     // [2,E]
    const int*   ei3    = (const int*)d_in[2];     // [2,E3]
    const int*   ai3    = (const int*)d_in[3];     // [2,A]
    const float* iso    = (const float*)d_in[4];
    const int*   batch  = (const int*)d_in[5];
    const int*   batch3 = (const int*)d_in[6];
    const float* Wx     = (const float*)d_in[7];
    const float* bx     = (const float*)d_in[8];
    const float* Wg     = (const float*)d_in[9];   // [L,EMB,EMB]
    const float* bg     = (const float*)d_in[10];  // [L,EMB]
    const float* gamma  = (const float*)d_in[11];
    const float* beta   = (const float*)d_in[12];
    const float* W3rel  = (const float*)d_in[13];
    const float* W3root = (const float*)d_in[14];
    const float* b3     = (const float*)d_in[15];
    const float* W4rel  = (const float*)d_in[16];
    const float* W4root = (const float*)d_in[17];
    const float* b4     = (const float*)d_in[18];
    const float* W0     = (const float*)d_in[19];
    const float* b0     = (const float*)d_in[20];
    const float* W1     = (const float*)d_in[21];
    const float* b1     = (const float*)d_in[22];
    const float* W2     = (const float*)d_in[23];
    const float* b2     = (const float*)d_in[24];
    const float* Wl     = (const float*)d_in[25];
    const float* bl     = (const float*)d_in[26];
    float* out = (float*)d_out;

    // ---------------- workspace layout (bytes) ----------------
    char* ws = (char*)d_ws;
    const size_t SZ_NODE = (size_t)NN * EMB * 4;           // 102,400,000
    float* h    = (float*)(ws + 0);
    float* hw   = (float*)(ws + SZ_NODE);
    float* agg  = (float*)(ws + 2 * SZ_NODE);
    float* dinv = (float*)(ws + 3 * SZ_NODE);              // NN floats
    float* enrm = (float*)(ws + 3 * SZ_NODE + 400128);     // EE floats
    char*  smal = ws + 3 * SZ_NODE + 400128 + (size_t)EE * 4;
    float* gsum = (float*)(smal);                          // G*EMB
    float* gcnt = (float*)(smal + 2048000);                // G
    float* mbuf = (float*)(smal + 2048000 + 8192);         // G*512
    float* m0b  = (float*)(smal + 2048000 + 8192 + 4096000);          // G*256
    float* m1b  = (float*)(smal + 2048000 + 8192 + 4096000 + 2048000);// G*128
    float* m2b  = (float*)(smal + 2048000 + 8192 + 4096000 + 3072000);// G*64

    // EFG-stage aliases (valid after node stage + pooling retire, stream order)
    float* xp    = hw;                                     // N3*EMB
    float* cnt3  = (float*)((char*)hw + 20480000);         // N3
    float* xc    = (float*)((char*)hw + 20480000 + 80128); // N3*(EMB+V)
    float* agg3  = agg;                                    // N3*(EMB+V)
    float* t3a   = (float*)((char*)agg + 37120000);        // N3*EMB
    float* t3b   = (float*)((char*)agg + 57600000);        // N3*EMB
    float* agg3b = h;                                      // N3*EMB
    float* t3c   = (float*)((char*)h + 20480000);          // N3*EMB
    float* h3b   = (float*)((char*)h + 40960000);          // N3*EMB

    auto fill = [&](float* p, float v, long long n) {
        fill_f32<<<(unsigned)((n + 255) / 256), 256, 0, stream>>>(p, v, n);
    };
    auto gemm = [&](const float* A, const float* W, const float* bias,
                    const float* Cadd, float* C, int M, int K, int Ncol, int relu) {
        unsigned blocks = (unsigned)((M / 16 + 3) / 4);
        if (Ncol == 256)
            gemm_wmma_f32_nc<256><<<blocks, 128, 0, stream>>>(A, W, bias, Cadd, C, M, K, relu);
        else if (Ncol == 128)
            gemm_wmma_f32_nc<128><<<blocks, 128, 0, stream>>>(A, W, bias, Cadd, C, M, K, relu);
        else
            gemm_wmma_f32_nc<64><<<blocks, 128, 0, stream>>>(A, W, bias, Cadd, C, M, K, relu);
    };

    const int* e_src = ei;            // edge_index[0]
    const int* e_dst = ei + EE;       // edge_index[1]
    const int* s3    = ei3;
    const int* d3    = ei3 + E3;
    const int* a_row = ai3;           // atom indices
    const int* a_col = ai3 + AA;      // cluster indices

    // --- degree / GCN norm ---
    fill(dinv, 1.0f, NN);                                  // self-loop contribution
    deg_accum<<<(EE + 255) / 256, 256, 0, stream>>>(e_dst, dinv, EE);
    rsqrt_inplace<<<(NN + 255) / 256, 256, 0, stream>>>(dinv, NN);
    edge_norm_k<<<(EE + 255) / 256, 256, 0, stream>>>(e_src, e_dst, dinv, enrm, EE);

    // --- h = x @ Wx + bx ---
    gemm(x, Wx, bx, nullptr, h, NN, FIN, EMB, 0);

    // --- 5 GCN layers ---
    const long long totN = (long long)NN * EMB;
    const long long totE = (long long)EE * (EMB / 4);
    for (int l = 0; l < LL; ++l) {
        fill(agg, 0.0f, totN);
        gemm(h, Wg + (size_t)l * EMB * EMB, nullptr, nullptr, hw, NN, EMB, EMB, 0);
        scatter_edges<<<(unsigned)((totE + 255) / 256), 256, 0, stream>>>(
            hw, e_src, e_dst, enrm, agg, totE, EMB / 4, EMB);
        layer_update<<<(unsigned)((totN + 255) / 256), 256, 0, stream>>>(
            h, hw, agg, dinv, bg + (size_t)l * EMB, gamma + (size_t)l * EMB,
            beta + (size_t)l * EMB, totN, EMB, (l < LL - 1) ? 1 : 0);
    }

    // --- x1 = seg_mean(h, batch, G) -> mbuf[:, 0:256] ---
    fill(gsum, 0.0f, (long long)GG * EMB);
    fill(gcnt, 0.0f, GG);
    seg_pool<<<(unsigned)(((long long)NN * (EMB / 4) + 255) / 256), 256, 0, stream>>>(
        h, batch, gsum, gcnt, (long long)NN * (EMB / 4), EMB / 4, EMB);
    mean_to<<<(unsigned)(((long long)GG * EMB + 255) / 256), 256, 0, stream>>>(
        mbuf, 2 * EMB, 0, gsum, gcnt, (long long)GG * EMB, EMB);

    // --- xp = seg_mean(h[a_row], a_col, N3); xc = [xp, iso] ---
    fill(xp, 0.0f, (long long)N3 * EMB);
    fill(cnt3, 0.0f, N3);
    seg_pool_gather<<<(unsigned)(((long long)AA * (EMB / 4) + 255) / 256), 256, 0, stream>>>(
        h, a_row, a_col, xp, cnt3, (long long)AA * (EMB / 4), EMB / 4, EMB);
    build_xc<<<(unsigned)(((long long)N3 * (EMB + VV) + 255) / 256), 256, 0, stream>>>(
        xp, cnt3, iso, xc, (long long)N3 * (EMB + VV));

    // --- GraphConv 1: h3 = relu( seg_sum(xc[s3],d3) @ W3rel + xc @ W3root + b3 ) ---
    const int FC = EMB + VV;                   // 464
    fill(agg3, 0.0f, (long long)N3 * FC);
    scatter_edges<<<(unsigned)(((long long)E3 * (FC / 4) + 255) / 256), 256, 0, stream>>>(
        xc, s3, d3, nullptr, agg3, (long long)E3 * (FC / 4), FC / 4, FC);
    gemm(agg3, W3rel, b3, nullptr, t3a, N3, FC, EMB, 0);
    gemm(xc, W3root, nullptr, t3a, t3b, N3, FC, EMB, 1);

    // --- GraphConv 2: h3b = relu( seg_sum(t3b[s3],d3) @ W4rel + t3b @ W4root + b4 ) ---
    fill(agg3b, 0.0f, (long long)N3 * EMB);
    scatter_edges<<<(unsigned)(((long long)E3 * (EMB / 4) + 255) / 256), 256, 0, stream>>>(
        t3b, s3, d3, nullptr, agg3b, (long long)E3 * (EMB / 4), EMB / 4, EMB);
    gemm(agg3b, W4rel, b4, nullptr, t3c, N3, EMB, EMB, 0);
    gemm(t3b, W4root, nullptr, t3c, h3b, N3, EMB, EMB, 1);

    // --- x3 = seg_mean(h3b, batch_3, G) -> mbuf[:, 256:512] ---
    fill(gsum, 0.0f, (long long)GG * EMB);
    fill(gcnt, 0.0f, GG);
    seg_pool<<<(unsigned)(((long long)N3 * (EMB / 4) + 255) / 256), 256, 0, stream>>>(
        h3b, batch3, gsum, gcnt, (long long)N3 * (EMB / 4), EMB / 4, EMB);
    mean_to<<<(unsigned)(((long long)GG * EMB + 255) / 256), 256, 0, stream>>>(
        mbuf, 2 * EMB, EMB, gsum, gcnt, (long long)GG * EMB, EMB);

    // --- readout MLP ---
    gemm(mbuf, W0, b0, nullptr, m0b, GG, 2 * EMB, EMB, 1);      // 2000x512 @ 512x256
    gemm(m0b, W1, b1, nullptr, m1b, GG, EMB, EMB / 2, 1);       // @ 256x128
    gemm(m1b, W2, b2, nullptr, m2b, GG, EMB / 2, EMB / 4, 1);   // @ 128x64
    final_linear<<<(GG + 255) / 256, 256, 0, stream>>>(m2b, Wl, bl, out, GG, EMB / 4);
}